// PatchICL_Level_53352083751494
// MI455X (gfx1250) — compile-verified
//
#include <hip/hip_runtime.h>
#include <stdint.h>

#define RES 512
#define PS  64
#define KP  512         // patches per batch
#define NB  16          // batch
#define PATCH_ELEMS (PS * PS)   // 4096 floats = 16KB

#if __has_builtin(__builtin_amdgcn_global_load_async_to_lds_b128)
#define HAVE_ASYNC_LDS 1
#else
#define HAVE_ASYNC_LDS 0
#endif

// Match the builtin's parameter type exactly: GCC-style int vector of 16B,
// qualified with the target address spaces (1 = global, 3 = LDS).
typedef int v4i __attribute__((vector_size(16)));
typedef __attribute__((address_space(1))) v4i GV4;
typedef __attribute__((address_space(3))) v4i LV4;

__device__ __forceinline__ void wait_asynccnt0() {
#if __has_builtin(__builtin_amdgcn_s_wait_asynccnt)
    __builtin_amdgcn_s_wait_asynccnt(0);
#else
    asm volatile("s_wait_asynccnt 0" ::: "memory");
#endif
}

// ---------------------------------------------------------------------------
// Kernel: one 64x64 patch per block (8 wave32's). Stage patch into LDS via
// async global->LDS DMA, then scatter with L2-resident f32 atomics. Counts use
// one packed u64 atomic per pixel-pair when 8B-aligned (w0 even).
// (Placed first in the file so the disasm snippet shows its body — need to
//  verify native global_atomic_add_f32 lowering, not a CAS loop.)
// ---------------------------------------------------------------------------
__global__ __launch_bounds__(256) void scatter_kernel(
    const float* __restrict__ logits, const int* __restrict__ coords,
    float* __restrict__ sums, uint32_t* __restrict__ counts) {

    const int pid = blockIdx.x;                  // 0 .. NB*KP-1
    const int b   = pid >> 9;                    // pid / 512
    const float* patch = logits + (size_t)pid * PATCH_ELEMS;
    const int h0 = coords[2 * pid + 0];
    const int w0 = coords[2 * pid + 1];

    float*    sbase = sums   + (size_t)b * RES * RES;
    uint32_t* cbase = counts + (size_t)b * RES * RES;

    __shared__ float tile[PATCH_ELEMS];          // 16 KB

    // gfx1250 prefetch hint for the streaming patch data
    __builtin_prefetch(patch, 0, 1);

#if HAVE_ASYNC_LDS
    // 256 lanes x 16B x 4 iters = 16KB patch staged by the async global->LDS path
    #pragma unroll
    for (int it = 0; it < 4; ++it) {
        const int v4 = it * 256 + threadIdx.x;   // float4 index
        __builtin_amdgcn_global_load_async_to_lds_b128(
            (GV4*)(patch + v4 * 4),
            (LV4*)(&tile[v4 * 4]),
            0, 0);
    }
    wait_asynccnt0();
    __syncthreads();
#else
    #pragma unroll
    for (int it = 0; it < 4; ++it) {
        const int v4 = it * 256 + threadIdx.x;
        ((float4*)tile)[v4] = ((const float4*)patch)[v4];
    }
    __syncthreads();
#endif

    // Each thread owns a 2-wide column pair; adjacent lanes -> contiguous
    // canvas addresses -> coalesced atomic traffic into one L2 sector.
    const int  cp      = threadIdx.x & 31;       // column pair 0..31
    const int  row0    = threadIdx.x >> 5;       // 0..7
    const bool aligned = ((w0 & 1) == 0);

    #pragma unroll
    for (int it = 0; it < 8; ++it) {
        const int row = row0 + it * 8;
        const int h   = h0 + row;
        const int w   = w0 + 2 * cp;

        const float v0 = tile[row * PS + 2 * cp + 0];
        const float v1 = tile[row * PS + 2 * cp + 1];

        float* srow = sbase + h * RES + w;
        __hip_atomic_fetch_add(srow + 0, v0, __ATOMIC_RELAXED, __HIP_MEMORY_SCOPE_AGENT);
        __hip_atomic_fetch_add(srow + 1, v1, __ATOMIC_RELAXED, __HIP_MEMORY_SCOPE_AGENT);

        uint32_t* crow = cbase + h * RES + w;
        if (aligned) {
            // +1 to two adjacent u32 counters in a single 64-bit atomic
            __hip_atomic_fetch_add((unsigned long long*)crow,
                                   0x0000000100000001ULL,
                                   __ATOMIC_RELAXED, __HIP_MEMORY_SCOPE_AGENT);
        } else {
            __hip_atomic_fetch_add(crow + 0, 1u, __ATOMIC_RELAXED, __HIP_MEMORY_SCOPE_AGENT);
            __hip_atomic_fetch_add(crow + 1, 1u, __ATOMIC_RELAXED, __HIP_MEMORY_SCOPE_AGENT);
        }
    }
}

// ---------------------------------------------------------------------------
// Kernel: zero the f32 sum accumulator (d_out) and u32 count accumulator (ws)
// ---------------------------------------------------------------------------
__global__ __launch_bounds__(256) void zero_kernel(float* __restrict__ sums,
                                                   uint32_t* __restrict__ counts,
                                                   int n4) {
    int i = blockIdx.x * 256 + threadIdx.x;
    if (i < n4) {
        ((float4*)sums)[i]  = make_float4(0.f, 0.f, 0.f, 0.f);
        ((uint4*)counts)[i] = make_uint4(0u, 0u, 0u, 0u);
    }
}

// ---------------------------------------------------------------------------
// Kernel: out = sum / max(count, 1)   (in place, float4/uint4 traffic)
// ---------------------------------------------------------------------------
__global__ __launch_bounds__(256) void normalize_kernel(float* __restrict__ out,
                                                        const uint32_t* __restrict__ counts,
                                                        int n4) {
    int i = blockIdx.x * 256 + threadIdx.x;
    if (i < n4) {
        float4 s = ((float4*)out)[i];
        uint4  c = ((const uint4*)counts)[i];
        s.x /= fmaxf((float)c.x, 1.0f);
        s.y /= fmaxf((float)c.y, 1.0f);
        s.z /= fmaxf((float)c.z, 1.0f);
        s.w /= fmaxf((float)c.w, 1.0f);
        ((float4*)out)[i] = s;
    }
}

extern "C" void kernel_launch(void* const* d_in, const int* in_sizes, int n_in,
                              void* d_out, int out_size, void* d_ws, size_t ws_size,
                              hipStream_t stream) {
    (void)in_sizes; (void)n_in; (void)out_size; (void)ws_size;

    const float* logits = (const float*)d_in[0];  // [16,512,1,64,64] f32
    const int*   coords = (const int*)d_in[1];    // [16,512,2] i32

    float*    sums   = (float*)d_out;             // [16,512,512] accumulate here
    uint32_t* counts = (uint32_t*)d_ws;           // 16 MB of scratch

    const int n  = NB * RES * RES;                // 4,194,304
    const int n4 = n / 4;                         // 1,048,576

    zero_kernel<<<n4 / 256, 256, 0, stream>>>(sums, counts, n4);
    scatter_kernel<<<NB * KP, 256, 0, stream>>>(logits, coords, sums, counts);
    normalize_kernel<<<n4 / 256, 256, 0, stream>>>(sums, counts, n4);
}